// TransformConv1d_39264591020709
// MI455X (gfx1250) — compile-verified
//
#include <hip/hip_runtime.h>

// TransformConv1d: out[b,o,n] = sum_{k,c} in[b,c,n] * w[o,k,c] * score[b,k,n]
// B=8, C_IN=C_OUT=16, K=4, N=524288, all fp32.
//
// Per (b, 16-wide n tile): out_tile = sum_k (W_k @ in_tile) .* s_k via
// V_WMMA_F32_16X16X4_F32 (16 chained WMMAs per tile, score folded into B
// with v_pk_mul_f32). Batch rides on blockIdx.y; loads are scalar-row-base +
// 32-bit voffset, stores/prefetches are pinned to the GVS saddr form via
// inline asm so no 64-bit per-lane address chains remain in the loop.

typedef __attribute__((ext_vector_type(2))) float v2f;
typedef __attribute__((ext_vector_type(8))) float v8f;

#define B_DIM  8
#define C_IN   16
#define C_OUT  16
#define K_TAPS 4

__global__ __launch_bounds__(256) void tconv1d_wmma_f32(
    const float* __restrict__ in,   // [B, C_IN, N]
    const float* __restrict__ wgt,  // [C_OUT, K, C_IN]
    const float* __restrict__ sc,   // [B, K, N]
    float* __restrict__ out,        // [B, C_OUT, N]
    int N, int tilesPerB)
{
    const int lane = (int)(threadIdx.x & 31u);
    const int half = lane >> 4;   // 0: lanes 0-15, 1: lanes 16-31
    const int col  = lane & 15;   // N within tile / M row for A-matrix

    // ---- Batch base pointers: uniform (SGPR) math, blockIdx.y = b.
    const int b = (int)blockIdx.y;
    const float* in_b  = in  + (size_t)b * C_IN   * (size_t)N;
    const float* sc_b  = sc  + (size_t)b * K_TAPS * (size_t)N;
    float*       out_b = out + (size_t)b * C_OUT  * (size_t)N;

    // ---- Scalar row base pointers (SGPR pairs).
    const float* inRow[8];   // input rows c = 4j + {0,1}; +2*half via voffset
#pragma unroll
    for (int j = 0; j < 4; ++j) {
        inRow[2 * j + 0] = in_b + (size_t)(4 * j + 0) * (size_t)N;
        inRow[2 * j + 1] = in_b + (size_t)(4 * j + 1) * (size_t)N;
    }
    const float* scRow[K_TAPS];
#pragma unroll
    for (int k = 0; k < K_TAPS; ++k)
        scRow[k] = sc_b + (size_t)k * (size_t)N;
    float* outRow[8];        // output rows o = v; +8*half via voffset
#pragma unroll
    for (int v = 0; v < 8; ++v)
        outRow[v] = out_b + (size_t)v * (size_t)N;

    // ---- Preload the whole weight tensor as WMMA A-fragments (16x4 f32).
    // A layout: lane = M (row o), K element = vgpr + 2*half.
    // A[k][j] covers c = 4j .. 4j+3 of W_k.
    v2f A[K_TAPS][4];
#pragma unroll
    for (int k = 0; k < K_TAPS; ++k) {
#pragma unroll
        for (int j = 0; j < 4; ++j) {
            const float* wp = wgt + (size_t)col * (K_TAPS * C_IN)
                                  + k * C_IN + 4 * j + 2 * half;
            A[k][j].x = wp[0];
            A[k][j].y = wp[1];
        }
    }

    // Wave index is wave-uniform: tell the compiler via readfirstlane so the
    // grid-stride loop runs on the SALU (s_cmp/s_cbranch, EXEC stays all-1s).
    const int wave = __builtin_amdgcn_readfirstlane(
        (int)((blockIdx.x * blockDim.x + threadIdx.x) >> 5));
    const int nWaves = (int)((gridDim.x * blockDim.x) >> 5);
    const unsigned strideEl = (unsigned)nWaves * 16u;   // grid-stride in n

    // Per-lane 32-bit element offsets; divergent parts folded in once.
    unsigned voff = (unsigned)wave * 16u + (unsigned)col;          // score
    const unsigned inHalf  = (unsigned)(2 * half) * (unsigned)N;   // B rows
    const unsigned outHalf = (unsigned)(8 * half) * (unsigned)N;   // D rows
    // Byte offset for prefetch (GVS form), one grid-stride tile ahead.
    unsigned pfB = (voff + inHalf + strideEl) * 4u;

    for (int t = wave; t < tilesPerB;
         t += nWaves, voff += strideEl, pfB += strideEl * 4u) {
        const unsigned vin  = voff + inHalf;
        const unsigned vout = voff + outHalf;

        // ---- B fragments: rows c = 4j + {0,1} + 2*half, column n0+col.
        // Each half-wave load is a contiguous 64B row -> fully coalesced.
        v2f Bf[4];
#pragma unroll
        for (int j = 0; j < 4; ++j) {
            Bf[j].x = inRow[2 * j + 0][vin];
            Bf[j].y = inRow[2 * j + 1][vin];
        }

        // ---- Per-lane score scalars (column n0+col), one per tap k.
        float s[K_TAPS];
#pragma unroll
        for (int k = 0; k < K_TAPS; ++k)
            s[k] = scRow[k][voff];

        // ---- Speculative prefetch of the next grid-stride tile's input
        // rows, pinned to GVS saddr form: one shared byte-offset VGPR, no
        // 64-bit address math, no LOADcnt. OOB translations drop silently.
#pragma unroll
        for (int j = 0; j < 4; ++j) {
            asm volatile("global_prefetch_b8 %0, %1 scope:SCOPE_SYS"
                         :: "v"(pfB), "s"(inRow[2 * j + 0]));
            asm volatile("global_prefetch_b8 %0, %1 scope:SCOPE_SYS"
                         :: "v"(pfB), "s"(inRow[2 * j + 1]));
        }

        // ---- 16 chained WMMAs: acc = A[k][j] x (Bf[j]*s[k]) + acc.
        // Accumulation flows through SRC2 (C), so no D->A/B RAW hazard.
        v8f acc = {0.f, 0.f, 0.f, 0.f, 0.f, 0.f, 0.f, 0.f};
#pragma unroll
        for (int k = 0; k < K_TAPS; ++k) {
#pragma unroll
            for (int j = 0; j < 4; ++j) {
                v2f bs;
                bs.x = Bf[j].x * s[k];
                bs.y = Bf[j].y * s[k];
                acc = __builtin_amdgcn_wmma_f32_16x16x4_f32(
                    /*neg_a=*/false, A[k][j],
                    /*neg_b=*/false, bs,
                    /*c_mod=*/(short)0, acc,
                    /*reuse_a=*/false, /*reuse_b=*/false);
            }
        }

        // ---- Store D: row o = vgpr + 8*half. Pinned to the saddr +
        // voffset(scale_offset) form: zero per-store address VALU. Store
        // source VGPRs may be reused after issue (no WAR hazard per ISA).
#pragma unroll
        for (int v = 0; v < 8; ++v) {
            asm volatile("global_store_b32 %0, %1, %2 scale_offset"
                         :: "v"(vout), "v"(acc[v]), "s"(outRow[v])
                         : "memory");
        }
    }
}

extern "C" void kernel_launch(void* const* d_in, const int* in_sizes, int n_in,
                              void* d_out, int out_size, void* d_ws, size_t ws_size,
                              hipStream_t stream)
{
    const float* in  = (const float*)d_in[0];  // [8,16,N]
    const float* wgt = (const float*)d_in[1];  // [16,4,16]
    const float* sc  = (const float*)d_in[2];  // [8,4,N]
    float*       out = (float*)d_out;          // [8,16,N]

    const long long N64 = (long long)in_sizes[0] / (B_DIM * C_IN);
    const int N = (int)N64;                    // 524288
    const int tilesPerB = N / 16;              // 32768 tiles per batch

    // 256 threads = 8 wave32 per block; blockIdx.y = batch.
    // 256 blocks/batch -> 2048 waves/batch -> 16 tiles per wave.
    int blocksX = 256;
    const int maxBlocksX = (tilesPerB + 7) / 8;  // >=1 tile per wave
    if (blocksX > maxBlocksX) blocksX = maxBlocksX;
    if (blocksX < 1) blocksX = 1;

    dim3 grid((unsigned)blocksX, B_DIM, 1);
    tconv1d_wmma_f32<<<grid, 256, 0, stream>>>(in, wgt, sc, out, N, tilesPerB);
}